// Grouped_KA_attention_16655883173903
// MI455X (gfx1250) — compile-verified
//
#include <hip/hip_runtime.h>
#include <hip/hip_bf16.h>
#include <math.h>

// ---------------------------------------------------------------------------
// Problem constants (from reference): B=32, H=8, P=16, D=32 -> N = PHO = 4096
// NUM_F=8, GROUPS=64 (group size 64), OUT_DIM=32.
// Fused GEMM: A[32 x 40960] * W^T[40960 x 4096]
//   k in [0, 32768): conv part, f = k>>12, n = k&4095,
//       A = sin(g_f * q)*ce_q[f,n] + sin(g_f * k)*ce_k[f,n]   (fq+fk fused)
//   k in [32768, 36864): silu(q),  weights = base_weight_q
//   k in [36864, 40960): silu(k),  weights = base_weight_k
// ---------------------------------------------------------------------------

#define BATCH   32
#define NDIM    4096
#define PHO_    4096
#define NUMF    8
#define KCONV   (NUMF * NDIM)        // 32768
#define KTOT    (KCONV + 2 * NDIM)   // 40960
#define KSEG    8192                 // K per segment (5 segments; seg 4 = base)
#define NSEG    5
#define KC      256                  // K staged in LDS per chunk
#define LDS_STRIDE 260               // pad 256 -> 260 floats: conflict-free b64

typedef __attribute__((ext_vector_type(2))) float v2f;
typedef __attribute__((ext_vector_type(4))) float v4f;
typedef __attribute__((ext_vector_type(8))) float v8f;

// ---------------------------------------------------------------------------
// Kernel 1: build fused activation matrix A[32][40960]
// ---------------------------------------------------------------------------
__global__ void build_A(const float* __restrict__ q, const float* __restrict__ kx,
                        const float* __restrict__ grid_f,
                        const float* __restrict__ coef_q,
                        const float* __restrict__ coef_k,
                        float* __restrict__ A) {
    int idx = blockIdx.x * blockDim.x + threadIdx.x;     // [0, 32*40960)
    int b = idx / KTOT;
    int kk = idx - b * KTOT;
    float val;
    if (kk < KCONV) {
        int f = kk >> 12;
        int n = kk & (NDIM - 1);
        int grp = n >> 6;                                // group size 64
        float g  = grid_f[f];
        float qv = q[b * NDIM + n];
        float kv = kx[b * NDIM + n];
        val = __sinf(g * qv) * coef_q[grp * NUMF + f]
            + __sinf(g * kv) * coef_k[grp * NUMF + f];
    } else if (kk < KCONV + NDIM) {
        float x = q[b * NDIM + (kk - KCONV)];
        val = x / (1.0f + __expf(-x));                   // silu
    } else {
        float x = kx[b * NDIM + (kk - KCONV - NDIM)];
        val = x / (1.0f + __expf(-x));
    }
    A[idx] = val;
}

// ---------------------------------------------------------------------------
// Kernel 2: WMMA GEMM. grid = (PHO/64 n-blocks, 5 k-segments), 256 thr/block.
// Wave w: mtile = w&1 (batch rows 0-15 / 16-31), ntile = w>>2? no: w>>1 ->
// 4 n-tiles of 16 outputs. A chunk staged in LDS, weights streamed from HBM.
// Writes fp32 partials [seg][32][4096] to workspace.
// ---------------------------------------------------------------------------
__global__ __launch_bounds__(256) void gemm_wmma(
    const float* __restrict__ A,
    const float* __restrict__ conv_wq,
    const float* __restrict__ base_wq,
    const float* __restrict__ base_wk,
    float* __restrict__ partial) {

    __shared__ float lds[BATCH * LDS_STRIDE];            // 33,280 bytes

    const int t     = threadIdx.x;
    const int wave  = t >> 5;
    const int lane  = t & 31;
    const int r     = lane & 15;                         // row within 16
    const int g     = lane >> 4;                         // lane half -> k pair
    const int mtile = wave & 1;
    const int ntile = wave >> 1;
    const int o     = blockIdx.x * 64 + ntile * 16 + r;  // output column
    const int kseg0 = blockIdx.y * KSEG;

    v8f acc = {};

    for (int c = 0; c < KSEG / KC; ++c) {
        const int kg = kseg0 + c * KC;

        __syncthreads();                                 // chunk c-1 consumed
        // Cooperative, coalesced stage of A[0:32][kg:kg+256] into LDS
        #pragma unroll
        for (int i = 0; i < 8; ++i) {
            int v   = i * 256 + t;                       // float4 index
            int row = v >> 6;
            int col = (v & 63) << 2;
            v4f d = *(const v4f*)(A + (size_t)row * KTOT + kg + col);
            *(v4f*)(&lds[row * LDS_STRIDE + col]) = d;
        }
        __syncthreads();

        // Weight row for this wave's output column o within this chunk.
        const float* wrow;
        if (kg < KCONV) {
            int f  = kg >> 12;
            int n0 = kg & (NDIM - 1);
            wrow = conv_wq + ((size_t)f * PHO_ + o) * NDIM + n0;
        } else if (kg < KCONV + NDIM) {
            wrow = base_wq + (size_t)o * NDIM + (kg - KCONV);
        } else {
            wrow = base_wk + (size_t)o * NDIM + (kg - KCONV - NDIM);
        }

        const float* arow = &lds[(mtile * 16 + r) * LDS_STRIDE + 2 * g];
        const float* wptr = wrow + 2 * g;

        // 64 x v_wmma_f32_16x16x4_f32 per chunk; one ds b64 + one global b64
        // per WMMA (fragment layout matches consecutive-k pairs per lane).
        #pragma unroll 8
        for (int kk = 0; kk < KC; kk += 4) {
            v2f a = *(const v2f*)(arow + kk);
            v2f w = *(const v2f*)(wptr + kk);
            acc = __builtin_amdgcn_wmma_f32_16x16x4_f32(
                      false, a, false, w, (short)0, acc, false, false);
        }
    }

    // D layout: lane l, elem j -> M = 8*(l>>4) + j, N = l&15
    float* pout = partial
        + ((size_t)blockIdx.y * BATCH + mtile * 16 + 8 * g) * PHO_ + o;
    #pragma unroll
    for (int j = 0; j < 8; ++j)
        pout[(size_t)j * PHO_] = acc[j];
}

// ---------------------------------------------------------------------------
// Kernel 3: combine partials + bias + scale, softmax over 32 consecutive
// outputs. One wave32 per softmax group (lane == element).
// ---------------------------------------------------------------------------
__global__ void softmax_out(const float* __restrict__ partial,
                            const float* __restrict__ conv_bq,
                            const float* __restrict__ scale_sp,
                            float* __restrict__ out) {
    int wid  = (blockIdx.x * blockDim.x + threadIdx.x) >> 5;  // [0, 32*128)
    int lane = threadIdx.x & 31;
    int b    = wid >> 7;
    int o    = (wid & 127) * 32 + lane;

    float conv = 0.0f;
    #pragma unroll
    for (int s = 0; s < 4; ++s)
        conv += partial[((size_t)s * BATCH + b) * PHO_ + o];
    float bias = 0.0f;
    #pragma unroll
    for (int f = 0; f < NUMF; ++f)
        bias += conv_bq[f * PHO_ + o];
    conv += 2.0f * bias;                 // bias appears once in fq and once in fk

    float sc = scale_sp[o];
    float y = partial[((size_t)4 * BATCH + b) * PHO_ + o]   // base_q + base_k
            + sc * sc * conv;                               // conv part * scale^2

    float m = y;
    #pragma unroll
    for (int off = 16; off > 0; off >>= 1)
        m = fmaxf(m, __shfl_xor(m, off, 32));
    float e = __expf(y - m);
    float sum = e;
    #pragma unroll
    for (int off = 16; off > 0; off >>= 1)
        sum += __shfl_xor(sum, off, 32);

    out[(size_t)b * PHO_ + o] = e / sum;
}

// ---------------------------------------------------------------------------
extern "C" void kernel_launch(void* const* d_in, const int* in_sizes, int n_in,
                              void* d_out, int out_size, void* d_ws, size_t ws_size,
                              hipStream_t stream) {
    const float* q    = (const float*)d_in[0];
    const float* k    = (const float*)d_in[1];
    const float* grd  = (const float*)d_in[2];
    const float* bwq  = (const float*)d_in[3];
    const float* bwk  = (const float*)d_in[4];
    const float* cq   = (const float*)d_in[5];
    const float* ck   = (const float*)d_in[6];
    const float* cwq  = (const float*)d_in[7];
    const float* cbq  = (const float*)d_in[8];
    const float* ssp  = (const float*)d_in[9];
    float* out = (float*)d_out;

    float* A       = (float*)d_ws;                   // 32*40960 fp32 = 5.25 MB
    float* partial = A + (size_t)BATCH * KTOT;       // 5*32*4096 fp32 = 2.6 MB

    build_A<<<dim3((BATCH * KTOT) / 256), dim3(256), 0, stream>>>(
        q, k, grd, cq, ck, A);
    gemm_wmma<<<dim3(PHO_ / 64, NSEG), dim3(256), 0, stream>>>(
        A, cwq, bwq, bwk, partial);
    softmax_out<<<dim3(512), dim3(256), 0, stream>>>(partial, cbq, ssp, out);
}